// Quantizer_4355096838566
// MI455X (gfx1250) — compile-verified
//
#include <hip/hip_runtime.h>

typedef float v2f __attribute__((ext_vector_type(2)));
typedef float v4f __attribute__((ext_vector_type(4)));
typedef float v8f __attribute__((ext_vector_type(8)));
typedef unsigned int v4u __attribute__((ext_vector_type(4)));
typedef int v4i __attribute__((ext_vector_type(4)));
typedef int v8i __attribute__((ext_vector_type(8)));

#define B_     16
#define C_     64
#define L_     4096
#define K_     1024
#define NV     (B_ * L_)       // 65536 vectors
#define NTILE  (NV / 16)       // 4096 tiles of 16 vectors
#define KT     64              // codebook rows per LDS chunk (double buffered)
#define NCHUNK (K_ / KT)       // 16
#define ROWP   68              // padded LDS row stride in floats (bank stride 4)
#define WAVES  4
#define TPW    2               // l-tiles per wave (B-fragment reuse)
#define TPB    (WAVES * 32)

#define USE_TDM 1              // flip to 0 to fall back to sync cooperative loads

// ---------------------------------------------------------------------------
// Kernel 1: codebook squared norms  w2[k] = sum_c cb[k][c]^2
// ---------------------------------------------------------------------------
__global__ void vq_w2_kernel(const float* __restrict__ cb, float* __restrict__ w2) {
    int k = blockIdx.x * blockDim.x + threadIdx.x;
    if (k < K_) {
        const float* row = cb + (size_t)k * C_;
        float s = 0.f;
#pragma unroll 8
        for (int c = 0; c < C_; ++c) s += row[c] * row[c];
        w2[k] = s;
    }
}

#if USE_TDM
// ---------------------------------------------------------------------------
// TDM: async DMA of one KT x 64 f32 codebook chunk into LDS, with hardware
// padding of 4 DWORDs every 64 DWORDs -> 68-float padded rows in LDS.
// D# layout per CDNA5 ISA ch.8 (2D tile, groups 2/3 disabled via zeros).
// This toolchain exposes the 6-arg builtin:
//   (uint32x4 g0, int32x8 g1, int32x4 g2, int32x4 g3, int32x8, i32 cpol)
// ---------------------------------------------------------------------------
__device__ __forceinline__ void tdm_load_chunk(const float* gsrc, float* ldsdst) {
    unsigned long long ga  = (unsigned long long)(uintptr_t)gsrc;
    unsigned int       lds = (unsigned int)(uintptr_t)ldsdst;  // addr[31:0] = LDS offset

    v4u g0;
    g0[0] = 1u;                                        // count=1 (valid user D#)
    g0[1] = lds;                                       // lds_addr (bytes)
    g0[2] = (unsigned int)(ga & 0xFFFFFFFFu);          // global_addr[31:0]
    g0[3] = (unsigned int)((ga >> 32) & 0x1FFFFFFu)    // global_addr[56:32]
          | (2u << 30);                                // type=2 ("image")

    v8i g1;
    g1[0] = (2 << 16)        // data_size = 4B
          | (1 << 20)        // pad_enable
          | (5 << 22)        // pad_interval: 64 DWORDs
          | (3 << 25);       // pad_amount: 4 DWORDs  -> ROWP = 68 floats
    g1[1] = (64 << 16);      // tensor_dim0[15:0] = 64 (bits 79:48 low half)
    g1[2] = (KT << 16);      // tensor_dim0 hi = 0 | tensor_dim1[15:0] = KT
    g1[3] = (64 << 16);      // tensor_dim1 hi = 0 | tile_dim0 = 64
    g1[4] = KT;              // tile_dim1 = KT, tile_dim2 = 0
    g1[5] = 64;              // tensor_dim0_stride[31:0] = 64 elements
    g1[6] = 0;               // stride hi / tensor_dim1_stride lo (unused for 2D)
    g1[7] = 0;

    v4i z4 = {0, 0, 0, 0};
    v8i z8 = {0, 0, 0, 0, 0, 0, 0, 0};
    __builtin_amdgcn_tensor_load_to_lds(g0, g1, z4, z4, z8, 0);
}
#endif

// ---------------------------------------------------------------------------
// Kernel 2: main VQ kernel. Each wave owns TWO 16-vector tiles; every LDS
// B-fragment feeds two v_wmma_f32_16x16x4_f32 chains. Codebook chunks are
// double-buffered in LDS via the Tensor Data Mover.
// ---------------------------------------------------------------------------
__global__ void __launch_bounds__(TPB)
vq_main_kernel(const float* __restrict__ x, const float* __restrict__ cb,
               const float* __restrict__ w2g, float* __restrict__ quant,
               int* __restrict__ idx_out, float* __restrict__ tile_loss) {
    __shared__ float lds_cb[2][KT * ROWP];
    __shared__ int   lds_kmin[WAVES * TPW * 16];

    const int tid  = threadIdx.x;
    const int wave = tid >> 5;
    const int lane = tid & 31;
    const int n    = lane & 15;   // k-column / m-row sub-id per ISA layout
    const int hc   = lane >> 4;   // half-wave selector

    const int tileBase = (blockIdx.x * WAVES + wave) * TPW;
    const int vb0      = tileBase * 16;       // 32 consecutive (b,l) vectors
    const int b        = vb0 >> 12;           // / L_
    const int l0       = vb0 & (L_ - 1);
    const float* xb    = x + (size_t)b * C_ * L_ + l0;

    // ---- Load two A strips (2 x 16 rows x C=64) ---------------------------
    v2f a0[16], a1[16];
    float x2p0 = 0.f, x2p1 = 0.f;
#pragma unroll
    for (int ci = 0; ci < 16; ++ci) {
        int cA = ci * 4 + hc * 2;
        float a0x = xb[(size_t)cA * L_ + n];
        float a0y = xb[(size_t)(cA + 1) * L_ + n];
        float a1x = xb[(size_t)cA * L_ + 16 + n];
        float a1y = xb[(size_t)(cA + 1) * L_ + 16 + n];
        a0[ci].x = a0x; a0[ci].y = a0y;
        a1[ci].x = a1x; a1[ci].y = a1y;
        x2p0 += a0x * a0x + a0y * a0y;
        x2p1 += a1x * a1x + a1y * a1y;
    }
    float x2f0 = x2p0 + __shfl_xor(x2p0, 16, 32);
    float x2f1 = x2p1 + __shfl_xor(x2p1, 16, 32);

    float minv0[8], minv1[8];
    int   mink0[8], mink1[8];
#pragma unroll
    for (int r = 0; r < 8; ++r) {
        minv0[r] = 3.4e38f; mink0[r] = 0;
        minv1[r] = 3.4e38f; mink1[r] = 0;
    }

#if USE_TDM
    if (wave == 0) tdm_load_chunk(cb, &lds_cb[0][0]);   // prologue: chunk 0
#endif

    for (int chunk = 0; chunk < NCHUNK; ++chunk) {
        const int cur   = chunk & 1;
        const int kbase = chunk * KT;
#if USE_TDM
        if (wave == 0) __builtin_amdgcn_s_wait_tensorcnt(0);
        __syncthreads();                                 // buf[cur] ready for all
        if (wave == 0 && (chunk + 1) < NCHUNK)
            tdm_load_chunk(cb + (size_t)(chunk + 1) * KT * C_, &lds_cb[cur ^ 1][0]);
#else
        __syncthreads();
        {   // sync cooperative load of chunk into buf[cur]
            const v4f* src = (const v4f*)(cb + (size_t)kbase * C_);
#pragma unroll
            for (int i = 0; i < (KT * C_ / 4) / TPB; ++i) {
                int e4   = tid + i * TPB;
                int row  = e4 >> 4;
                int col4 = e4 & 15;
                *(v4f*)&lds_cb[cur][row * ROWP + col4 * 4] = src[e4];
            }
        }
        __syncthreads();
#endif
        const float* lcb = &lds_cb[cur][0];

#pragma unroll
        for (int kt = 0; kt < KT / 16; ++kt) {
            const int rb = kt * 16;
            const int k0 = kbase + rb;
            v8f d0 = {0.f, 0.f, 0.f, 0.f, 0.f, 0.f, 0.f, 0.f};
            v8f d1 = {0.f, 0.f, 0.f, 0.f, 0.f, 0.f, 0.f, 0.f};
#pragma unroll
            for (int ci = 0; ci < 16; ++ci) {
                int cB = ci * 4 + hc * 2;
                v2f bv;
                bv.x = lcb[(rb + n) * ROWP + cB];
                bv.y = lcb[(rb + n) * ROWP + cB + 1];
                d0 = __builtin_amdgcn_wmma_f32_16x16x4_f32(
                        false, a0[ci], false, bv, (short)0, d0, false, false);
                d1 = __builtin_amdgcn_wmma_f32_16x16x4_f32(
                        false, a1[ci], false, bv, (short)0, d1, false, false);
            }
            const float w2v = w2g[k0 + n];
            const int   kk  = k0 + n;
#pragma unroll
            for (int r = 0; r < 8; ++r) {
                float s0 = w2v - 2.0f * d0[r];
                bool  t0 = s0 < minv0[r];
                minv0[r] = t0 ? s0 : minv0[r];
                mink0[r] = t0 ? kk : mink0[r];
                float s1 = w2v - 2.0f * d1[r];
                bool  t1 = s1 < minv1[r];
                minv1[r] = t1 ? s1 : minv1[r];
                mink1[r] = t1 ? kk : mink1[r];
            }
        }
        __syncthreads();      // all reads of buf[cur] done before it is re-filled
    }

    // ---- Cross-lane argmin within each 16-lane group ----------------------
#pragma unroll
    for (int r = 0; r < 8; ++r) {
#pragma unroll
        for (int off = 1; off < 16; off <<= 1) {
            float ov0 = __shfl_xor(minv0[r], off, 16);
            int   ok0 = __shfl_xor(mink0[r], off, 16);
            bool  t0  = (ov0 < minv0[r]) || (ov0 == minv0[r] && ok0 < mink0[r]);
            minv0[r] = t0 ? ov0 : minv0[r];
            mink0[r] = t0 ? ok0 : mink0[r];
            float ov1 = __shfl_xor(minv1[r], off, 16);
            int   ok1 = __shfl_xor(mink1[r], off, 16);
            bool  t1  = (ov1 < minv1[r]) || (ov1 == minv1[r] && ok1 < mink1[r]);
            minv1[r] = t1 ? ov1 : minv1[r];
            mink1[r] = t1 ? ok1 : mink1[r];
        }
    }

    // ---- Emit indices + tile loss partials (d2min = x2 + min(w2-2xw)) -----
    float lsum0 = 0.f, lsum1 = 0.f;
#pragma unroll
    for (int r = 0; r < 8; ++r) {
        int   mrow = r + 8 * hc;                 // D VGPR r -> M row per half
        float x2m0 = __shfl(x2f0, mrow, 16);
        float x2m1 = __shfl(x2f1, mrow, 16);
        if (n == 0) {
            idx_out[vb0 + mrow]      = mink0[r];
            idx_out[vb0 + 16 + mrow] = mink1[r];
            lds_kmin[(wave * TPW + 0) * 16 + mrow] = mink0[r];
            lds_kmin[(wave * TPW + 1) * 16 + mrow] = mink1[r];
            lsum0 += x2m0 + minv0[r];
            lsum1 += x2m1 + minv1[r];
        }
    }
    float tot0 = lsum0 + __shfl_xor(lsum0, 16, 32);
    float tot1 = lsum1 + __shfl_xor(lsum1, 16, 32);
    if (lane == 0) {
        tile_loss[tileBase]     = tot0;
        tile_loss[tileBase + 1] = tot1;
    }

    __syncthreads();

    // ---- Write quant_out = codebook[kmin] in (B, C, L) layout -------------
    const int m = n;
    float* ob = quant + (size_t)b * C_ * L_ + l0;
#pragma unroll
    for (int t = 0; t < TPW; ++t) {
        const int    km   = lds_kmin[(wave * TPW + t) * 16 + m];
        const float* crow = cb + (size_t)km * C_;
#pragma unroll
        for (int c0 = 0; c0 < C_; c0 += 2) {
            int c = c0 + hc;                 // halves cover even/odd c
            ob[(size_t)c * L_ + t * 16 + m] = crow[c];
        }
    }
}

// ---------------------------------------------------------------------------
// Kernel 3: deterministic reduce of tile partials -> both scalar losses
// ---------------------------------------------------------------------------
__global__ void vq_loss_kernel(const float* __restrict__ tile_loss,
                               float* __restrict__ out_losses) {
    __shared__ float sm[256];
    float s = 0.f;
    for (int i = threadIdx.x; i < NTILE; i += 256) s += tile_loss[i];
    sm[threadIdx.x] = s;
    __syncthreads();
    for (int st = 128; st > 0; st >>= 1) {
        if (threadIdx.x < st) sm[threadIdx.x] += sm[threadIdx.x + st];
        __syncthreads();
    }
    if (threadIdx.x == 0) {
        float loss = sm[0] / (float)((size_t)B_ * L_ * C_);
        out_losses[0] = loss;   // codebook_loss
        out_losses[1] = loss;   // commitment_loss (identical value)
    }
}

// ---------------------------------------------------------------------------
extern "C" void kernel_launch(void* const* d_in, const int* in_sizes, int n_in,
                              void* d_out, int out_size, void* d_ws, size_t ws_size,
                              hipStream_t stream) {
    const float* x  = (const float*)d_in[0];   // (B, C, L) f32
    const float* cb = (const float*)d_in[1];   // (K, C)   f32

    float* ws = (float*)d_ws;
    float* w2 = ws;            // K_ floats
    float* tl = ws + K_;       // NTILE floats

    float* quant  = (float*)d_out;                         // B*C*L floats
    float* losses = (float*)d_out + (size_t)B_ * C_ * L_;  // 2 floats
    int*   idx    = (int*)(losses + 2);                    // B*L int32

    vq_w2_kernel<<<(K_ + 255) / 256, 256, 0, stream>>>(cb, w2);
    vq_main_kernel<<<NTILE / (WAVES * TPW), TPB, 0, stream>>>(x, cb, w2, quant, idx, tl);
    vq_loss_kernel<<<1, 256, 0, stream>>>(tl, losses);
}